// ClusterQuantizer_26886495273681
// MI455X (gfx1250) — compile-verified
//
#include <hip/hip_runtime.h>

typedef __attribute__((ext_vector_type(2))) float v2f;
typedef __attribute__((ext_vector_type(8))) float v8f;

#define VQ_D      256
#define VQ_K      1024
#define VQ_P      65536
#define VQ_TILES  64            // K / 16
#define VQ_DSTEPS 64            // D / 4
#define TILE_STRIDE 260         // 256 + 4 pad -> conflict-free ds_read_b64

// ---------------------------------------------------------------------------
// Kernel A: e2[k] = ||codebook_k||^2 ; zero counts + loss accumulator
// ---------------------------------------------------------------------------
__global__ __launch_bounds__(256) void vq_prep_kernel(
    const float* __restrict__ cb, float* __restrict__ e2,
    unsigned int* __restrict__ counts, float* __restrict__ lossAccum)
{
    int k = blockIdx.x * 256 + threadIdx.x;
    if (k < VQ_K) {
        float s = 0.0f;
        #pragma unroll 8
        for (int d = 0; d < VQ_D; ++d) {
            float c = cb[k * VQ_D + d];
            s += c * c;
        }
        e2[k] = s;
        counts[k] = 0u;
    }
    if (k == 0) *lossAccum = 0.0f;
}

// ---------------------------------------------------------------------------
// Kernel B: WMMA f32 scores  score[p,k] = e2[k] - 2 * dot(z_p, e_k),
// running argmin per point, cross-lane min reduce, write indices + histogram.
// grid = 512 blocks x 256 threads (8 waves); each wave: 16 points vs all K.
// ---------------------------------------------------------------------------
__global__ __launch_bounds__(256) void vq_argmin_kernel(
    const float* __restrict__ z, const float* __restrict__ cb,
    const float* __restrict__ e2, int* __restrict__ idx_out,
    float* __restrict__ idx_f_out, unsigned int* __restrict__ counts)
{
    __shared__ float tile[16 * TILE_STRIDE];   // one 16x256 codebook tile (padded)

    const int lane = threadIdx.x & 31;
    const int wave = threadIdx.x >> 5;
    const int pb = blockIdx.x * 8 + wave;      // 16-point block id
    const int rowBase = pb * 16;
    const int m   = lane & 15;                 // point row (A) / code col (B,C)
    const int off = (lane >> 4) * 2;           // K sub-offset: 0 (low half) / 2 (high)

    // --- preload A fragments: z rows rowBase..rowBase+15, all of D, in regs ---
    v2f a[VQ_DSTEPS];
    const float* zrow = z + (size_t)(rowBase + m) * VQ_D + off;
    #pragma unroll
    for (int d = 0; d < VQ_DSTEPS; ++d)
        a[d] = *(const v2f*)(zrow + d * 4);

    float best[8];
    int   bidx[8];
    #pragma unroll
    for (int v = 0; v < 8; ++v) { best[v] = 3.0e38f; bidx[v] = 0; }

    for (int nt = 0; nt < VQ_TILES; ++nt) {
        // cooperative load: 16 codebook rows -> LDS (padded rows)
        const float* src = cb + (size_t)nt * 16 * VQ_D;
        #pragma unroll
        for (int i = 0; i < 4; ++i) {
            int f   = i * 1024 + threadIdx.x * 4;      // flat element id, 16B aligned
            int row = f >> 8;
            int col = f & 255;
            *(float4*)(&tile[row * TILE_STRIDE + col]) = *(const float4*)(src + f);
        }
        __syncthreads();

        v8f acc = {};
        const float* brow = &tile[m * TILE_STRIDE + off];
        #pragma unroll
        for (int d = 0; d < VQ_DSTEPS; ++d) {
            v2f b = *(const v2f*)(brow + d * 4);
            acc = __builtin_amdgcn_wmma_f32_16x16x4_f32(
                false, a[d], false, b, (short)0, acc, false, false);
        }

        float e2n = e2[nt * 16 + m];
        #pragma unroll
        for (int v = 0; v < 8; ++v) {
            float s = e2n - 2.0f * acc[v];             // lane holds (M=v(+8), N=m)
            if (s < best[v]) { best[v] = s; bidx[v] = nt * 16 + m; }
        }
        __syncthreads();
    }

    // min-reduce across the 16 lanes holding the same matrix row M
    #pragma unroll
    for (int v = 0; v < 8; ++v) {
        #pragma unroll
        for (int msk = 1; msk < 16; msk <<= 1) {
            float os = __shfl_xor(best[v], msk, 32);
            int   oi = __shfl_xor(bidx[v], msk, 32);
            if (os < best[v] || (os == best[v] && oi < bidx[v])) {
                best[v] = os; bidx[v] = oi;
            }
        }
        if (m == v) {  // two writer lanes: lane v (M=v), lane 16+v (M=v+8)
            int row = rowBase + v + (lane >> 4) * 8;
            int bi  = bidx[v];
            idx_out[row]   = bi;
            idx_f_out[row] = (float)bi;
            atomicAdd(&counts[bi], 1u);
        }
    }
}

// ---------------------------------------------------------------------------
// Kernel C: z_q gather + sum((z_q - z)^2). 4 points per 256-thread block,
// 64 threads (float4 each) per point -> fully coalesced 1KB rows.
// ---------------------------------------------------------------------------
__global__ __launch_bounds__(256) void vq_gather_kernel(
    const float* __restrict__ z, const float* __restrict__ cb,
    const int* __restrict__ idx, float* __restrict__ zq,
    float* __restrict__ lossAccum)
{
    __shared__ float red[256];
    const int t = threadIdx.x & 63;
    const int p = blockIdx.x * 4 + (threadIdx.x >> 6);
    const int bi = idx[p];

    const float4 zv = *(const float4*)(z  + (size_t)p  * VQ_D + t * 4);
    const float4 cv = *(const float4*)(cb + (size_t)bi * VQ_D + t * 4);
    *(float4*)(zq + (size_t)p * VQ_D + t * 4) = cv;

    float dx = cv.x - zv.x, dy = cv.y - zv.y;
    float dz = cv.z - zv.z, dw = cv.w - zv.w;
    red[threadIdx.x] = dx * dx + dy * dy + dz * dz + dw * dw;
    __syncthreads();
    for (int ofs = 128; ofs > 0; ofs >>= 1) {
        if (threadIdx.x < (unsigned)ofs) red[threadIdx.x] += red[threadIdx.x + ofs];
        __syncthreads();
    }
    if (threadIdx.x == 0) atomicAdd(lossAccum, red[0]);
}

// ---------------------------------------------------------------------------
// Kernel D: perplexity + loss scalars
// out_scalars: [commitment, codebook, cluster, perplexity]
// ---------------------------------------------------------------------------
__global__ __launch_bounds__(1024) void vq_final_kernel(
    const unsigned int* __restrict__ counts, const float* __restrict__ lossAccum,
    float* __restrict__ out_scalars)
{
    __shared__ float red[1024];
    const int k = threadIdx.x;
    float c = (float)counts[k];
    float p = c / (65536.0f + 1e-5f);      // counts.sum() == P exactly
    red[k] = p * logf(p + 1e-5f);
    __syncthreads();
    for (int ofs = 512; ofs > 0; ofs >>= 1) {
        if (k < ofs) red[k] += red[k + ofs];
        __syncthreads();
    }
    if (k == 0) {
        float perp = expf(-red[0]);
        float mse  = *lossAccum / 16777216.0f;   // P * D
        out_scalars[0] = mse;                    // commitment_loss
        out_scalars[1] = mse;                    // codebook_loss
        out_scalars[2] = mse * 1.25f;            // cluster_loss = cb + 0.25*commit
        out_scalars[3] = perp;                   // perplexity
    }
}

// ---------------------------------------------------------------------------
extern "C" void kernel_launch(void* const* d_in, const int* in_sizes, int n_in,
                              void* d_out, int out_size, void* d_ws, size_t ws_size,
                              hipStream_t stream)
{
    const float* z  = (const float*)d_in[0];   // [16,64,64,256]
    const float* cb = (const float*)d_in[1];   // [1024,256]
    float* out = (float*)d_out;

    // d_out layout (return-tuple order, flattened):
    float* zq_out     = out;                   // 16777216
    float* scalars    = out + 16777216;        // 4 scalars
    float* idx_f_out  = out + 16777220;        // 65536 indices as float

    // workspace layout
    char* ws = (char*)d_ws;
    float*        e2        = (float*)(ws + 0);            //  4 KB
    unsigned int* counts    = (unsigned int*)(ws + 4096);  //  4 KB
    float*        lossAccum = (float*)(ws + 8192);         //  4 B
    int*          idx_ws    = (int*)(ws + 12288);          // 256 KB

    vq_prep_kernel<<<4, 256, 0, stream>>>(cb, e2, counts, lossAccum);
    vq_argmin_kernel<<<VQ_P / (16 * 8), 256, 0, stream>>>(z, cb, e2, idx_ws,
                                                          idx_f_out, counts);
    vq_gather_kernel<<<VQ_P / 4, 256, 0, stream>>>(z, cb, idx_ws, zq_out, lossAccum);
    vq_final_kernel<<<1, 1024, 0, stream>>>(counts, lossAccum, scalars);
}